// SimpleGAT_11295763989050
// MI455X (gfx1250) — compile-verified
//
#include <hip/hip_runtime.h>

typedef __attribute__((ext_vector_type(2))) float v2f;
typedef __attribute__((ext_vector_type(8))) float v8f;

#define IN_DIM    128
#define H1        4
#define CDIM      64
#define HC1       256          // H1*CDIM
#define NEG_SLOPE 0.2f
#define TB        256          // threads per block (8 wave32)

// ---------------------------------------------------------------------------
// helpers
// ---------------------------------------------------------------------------
__device__ __forceinline__ void edge_nodes(const int* __restrict__ ei, int E,
                                           int i, int& s, int& d) {
  if (i < E) { s = ei[i]; d = ei[E + i]; }
  else       { s = i - E; d = i - E; }     // self loops appended
}

// order-preserving float <-> uint encoding (for atomicMax on floats)
__device__ __forceinline__ unsigned f_enc(float f) {
  unsigned u = __float_as_uint(f);
  return (u & 0x80000000u) ? ~u : (u | 0x80000000u);
}
__device__ __forceinline__ float f_dec(unsigned e) {
  return __uint_as_float((e & 0x80000000u) ? (e ^ 0x80000000u) : ~e);
}
#define ENC_NEG_INF 0x007FFFFFu   // f_dec(ENC_NEG_INF) == -inf

__device__ __forceinline__ float leaky(float e) {
  return e > 0.0f ? e : NEG_SLOPE * e;
}

// ---------------------------------------------------------------------------
// fills
// ---------------------------------------------------------------------------
__global__ void fill_f32(float* __restrict__ p, float v, int n) {
  int t = blockIdx.x * blockDim.x + threadIdx.x;
  if (t < n) p[t] = v;
}
__global__ void fill_u32(unsigned* __restrict__ p, unsigned v, int n) {
  int t = blockIdx.x * blockDim.x + threadIdx.x;
  if (t < n) p[t] = v;
}

// ---------------------------------------------------------------------------
// fp32 WMMA GEMM: C[M,Nn] = A[M,K] @ B[K,Nn]   (M%16==0, Nn%16==0, K%4==0)
// one wave32 computes one 16x16 output tile via V_WMMA_F32_16X16X4_F32
// ---------------------------------------------------------------------------
__global__ void gat_gemm_f32(const float* __restrict__ A, const float* __restrict__ B,
                             float* __restrict__ C, int M, int Nn, int K) {
  int wave = (blockIdx.x * blockDim.x + threadIdx.x) >> 5;
  int lane = threadIdx.x & 31;
  int ntiles = Nn >> 4;
  int tm = wave / ntiles;
  int tn = wave - tm * ntiles;
  if (tm >= (M >> 4)) return;                 // wave-uniform: EXEC stays all-1s

  int half = lane >> 4;                       // 0: K pair {0,1}, 1: K pair {2,3}
  int l16  = lane & 15;
  const float* Arow = A + (size_t)((tm << 4) + l16) * K;    // A: lane%16 = M row
  const float* Bcol = B + ((tn << 4) + l16);                // B: lane%16 = N col

  v8f acc = {};
  for (int k0 = 0; k0 < K; k0 += 4) {
    int ka = k0 + 2 * half;
    v2f av, bv;
    av.x = Arow[ka];
    av.y = Arow[ka + 1];
    bv.x = Bcol[(size_t)ka * Nn];
    bv.y = Bcol[(size_t)(ka + 1) * Nn];
    acc = __builtin_amdgcn_wmma_f32_16x16x4_f32(false, av, false, bv,
                                                (short)0, acc, false, false);
  }
  // C/D layout: vgpr i holds M = i + 8*half, N = lane%16
  float* Cp = C + (size_t)((tm << 4) + 8 * half) * Nn + (tn << 4) + l16;
#pragma unroll
  for (int i = 0; i < 8; ++i) Cp[(size_t)i * Nn] = acc[i];
}

// ---------------------------------------------------------------------------
// per-node attention scores: a_s[n,h] = <hx[n,h,:], att_src[h,:]>, same for dst
// ---------------------------------------------------------------------------
__global__ void gat_scores(const float* __restrict__ hx,
                           const float* __restrict__ att_s,
                           const float* __restrict__ att_d,
                           float* __restrict__ a_s, float* __restrict__ a_d,
                           int N, int H) {
  int t = blockIdx.x * blockDim.x + threadIdx.x;
  if (t >= N * H) return;
  int n = t / H, h = t - n * H;
  const float4* hv = (const float4*)(hx + (size_t)n * H * CDIM + (size_t)h * CDIM);
  const float4* sv = (const float4*)(att_s + (size_t)h * CDIM);
  const float4* dv = (const float4*)(att_d + (size_t)h * CDIM);
  float ss = 0.0f, dd = 0.0f;
#pragma unroll
  for (int i = 0; i < CDIM / 4; ++i) {
    float4 x = hv[i], a = sv[i], b = dv[i];
    ss += x.x * a.x + x.y * a.y + x.z * a.z + x.w * a.w;
    dd += x.x * b.x + x.y * b.y + x.z * b.z + x.w * b.w;
  }
  a_s[t] = ss;
  a_d[t] = dd;
}

// ---------------------------------------------------------------------------
// segment max of leaky(a_s[src]+a_d[dst]) over dst
// ---------------------------------------------------------------------------
__global__ void gat_edge_max(const int* __restrict__ ei, int E, int Etot, int H,
                             const float* __restrict__ a_s,
                             const float* __restrict__ a_d,
                             unsigned* __restrict__ menc) {
  int t = blockIdx.x * blockDim.x + threadIdx.x;
  if (t >= Etot * H) return;
  int i = t / H, h = t - i * H;
  int s, d; edge_nodes(ei, E, i, s, d);
  float e = leaky(a_s[s * H + h] + a_d[d * H + h]);
  atomicMax(&menc[d * H + h], f_enc(e));
}

// segment sum of exp(e - m[dst])
__global__ void gat_edge_sum(const int* __restrict__ ei, int E, int Etot, int H,
                             const float* __restrict__ a_s,
                             const float* __restrict__ a_d,
                             const unsigned* __restrict__ menc,
                             float* __restrict__ ssum) {
  int t = blockIdx.x * blockDim.x + threadIdx.x;
  if (t >= Etot * H) return;
  int i = t / H, h = t - i * H;
  int s, d; edge_nodes(ei, E, i, s, d);
  float e = leaky(a_s[s * H + h] + a_d[d * H + h]);
  float p = expf(e - f_dec(menc[d * H + h]));
  atomicAdd(&ssum[d * H + h], p);
}

// out[dst] += alpha * hx[src]; H*CDIM/4 threads per edge, float4 granularity
__global__ void gat_edge_agg(const int* __restrict__ ei, int E, int Etot, int H,
                             const float* __restrict__ a_s,
                             const float* __restrict__ a_d,
                             const unsigned* __restrict__ menc,
                             const float* __restrict__ ssum,
                             const float* __restrict__ hx,
                             float* __restrict__ out) {
  int tpe = (H * CDIM) >> 2;                 // float4 chunks per edge
  long long t = (long long)blockIdx.x * blockDim.x + threadIdx.x;
  if (t >= (long long)Etot * tpe) return;
  int i = (int)(t / tpe);
  int j = (int)(t - (long long)i * tpe);     // float4 index within row
  int h = (j << 2) / CDIM;
  int s, d; edge_nodes(ei, E, i, s, d);
  float e = leaky(a_s[s * H + h] + a_d[d * H + h]);
  float alpha = expf(e - f_dec(menc[d * H + h])) / (ssum[d * H + h] + 1e-16f);
  float4 v = ((const float4*)(hx + (size_t)s * H * CDIM))[j];
  float* o = out + (size_t)d * H * CDIM + ((size_t)j << 2);
  atomicAdd(o + 0, v.x * alpha);
  atomicAdd(o + 1, v.y * alpha);
  atomicAdd(o + 2, v.z * alpha);
  atomicAdd(o + 3, v.w * alpha);
}

// out = elu(out + bias)   (layer-1 epilogue)
__global__ void gat_bias_elu(float* __restrict__ out, const float* __restrict__ b,
                             int n, int F) {
  int t = blockIdx.x * blockDim.x + threadIdx.x;
  if (t >= n) return;
  float v = out[t] + b[t % F];
  out[t] = v > 0.0f ? v : (expf(v) - 1.0f);
}

// out += bias             (layer-2 epilogue)
__global__ void gat_bias_add(float* __restrict__ out, const float* __restrict__ b,
                             int n, int F) {
  int t = blockIdx.x * blockDim.x + threadIdx.x;
  if (t >= n) return;
  out[t] += b[t % F];
}

// ---------------------------------------------------------------------------
static inline int gblk(long long n) { return (int)((n + TB - 1) / TB); }

extern "C" void kernel_launch(void* const* d_in, const int* in_sizes, int n_in,
                              void* d_out, int out_size, void* d_ws, size_t ws_size,
                              hipStream_t stream) {
  const float* x        = (const float*)d_in[0];
  const int*   ei       = (const int*)  d_in[1];
  const float* W1       = (const float*)d_in[2];
  const float* att_src1 = (const float*)d_in[3];
  const float* att_dst1 = (const float*)d_in[4];
  const float* b1       = (const float*)d_in[5];
  const float* W2       = (const float*)d_in[6];
  const float* att_src2 = (const float*)d_in[7];
  const float* att_dst2 = (const float*)d_in[8];
  const float* b2       = (const float*)d_in[9];

  const int N    = in_sizes[0] / IN_DIM;     // 50000
  const int E    = in_sizes[1] / 2;          // 800000
  const int Etot = E + N;                    // + self loops
  float* out = (float*)d_out;

  // ---- workspace carve (floats) ----
  char* w = (char*)d_ws;
  size_t off = 0;
  auto carve = [&](size_t nfloat) { void* p = w + off; off += nfloat * 4; return p; };
  float*    hx1  = (float*)   carve((size_t)N * HC1);
  float*    out1 = (float*)   carve((size_t)N * HC1);
  float*    hx2  = (float*)   carve((size_t)N * CDIM);
  float*    as1  = (float*)   carve((size_t)N * H1);
  float*    ad1  = (float*)   carve((size_t)N * H1);
  float*    s1   = (float*)   carve((size_t)N * H1);
  unsigned* m1   = (unsigned*)carve((size_t)N * H1);
  float*    as2  = (float*)   carve((size_t)N);
  float*    ad2  = (float*)   carve((size_t)N);
  float*    s2   = (float*)   carve((size_t)N);
  unsigned* m2   = (unsigned*)carve((size_t)N);

  // ================= layer 1 =================
  // hx1 = x @ W1   (M=N, Nn=256, K=128)  -- WMMA fp32
  {
    long long waves = (long long)(N >> 4) * (HC1 >> 4);
    gat_gemm_f32<<<gblk(waves * 32), TB, 0, stream>>>(x, W1, hx1, N, HC1, IN_DIM);
  }
  gat_scores<<<gblk((long long)N * H1), TB, 0, stream>>>(hx1, att_src1, att_dst1,
                                                         as1, ad1, N, H1);
  fill_u32<<<gblk((long long)N * H1), TB, 0, stream>>>(m1, ENC_NEG_INF, N * H1);
  fill_f32<<<gblk((long long)N * H1), TB, 0, stream>>>(s1, 0.0f, N * H1);
  fill_f32<<<gblk((long long)N * HC1), TB, 0, stream>>>(out1, 0.0f, N * HC1);

  gat_edge_max<<<gblk((long long)Etot * H1), TB, 0, stream>>>(ei, E, Etot, H1,
                                                              as1, ad1, m1);
  gat_edge_sum<<<gblk((long long)Etot * H1), TB, 0, stream>>>(ei, E, Etot, H1,
                                                              as1, ad1, m1, s1);
  gat_edge_agg<<<gblk((long long)Etot * (HC1 >> 2)), TB, 0, stream>>>(
      ei, E, Etot, H1, as1, ad1, m1, s1, hx1, out1);
  gat_bias_elu<<<gblk((long long)N * HC1), TB, 0, stream>>>(out1, b1, N * HC1, HC1);

  // ================= layer 2 =================
  // hx2 = out1 @ W2   (M=N, Nn=64, K=256)  -- WMMA fp32
  {
    long long waves = (long long)(N >> 4) * (CDIM >> 4);
    gat_gemm_f32<<<gblk(waves * 32), TB, 0, stream>>>(out1, W2, hx2, N, CDIM, HC1);
  }
  gat_scores<<<gblk((long long)N), TB, 0, stream>>>(hx2, att_src2, att_dst2,
                                                    as2, ad2, N, 1);
  fill_u32<<<gblk((long long)N), TB, 0, stream>>>(m2, ENC_NEG_INF, N);
  fill_f32<<<gblk((long long)N), TB, 0, stream>>>(s2, 0.0f, N);
  fill_f32<<<gblk((long long)N * CDIM), TB, 0, stream>>>(out, 0.0f, N * CDIM);

  gat_edge_max<<<gblk((long long)Etot), TB, 0, stream>>>(ei, E, Etot, 1,
                                                         as2, ad2, m2);
  gat_edge_sum<<<gblk((long long)Etot), TB, 0, stream>>>(ei, E, Etot, 1,
                                                         as2, ad2, m2, s2);
  gat_edge_agg<<<gblk((long long)Etot * (CDIM >> 2)), TB, 0, stream>>>(
      ei, E, Etot, 1, as2, ad2, m2, s2, hx2, out);
  gat_bias_add<<<gblk((long long)N * CDIM), TB, 0, stream>>>(out, b2, N * CDIM, CDIM);
}